// depthwise_conv_17128329576861
// MI455X (gfx1250) — compile-verified
//
#include <hip/hip_runtime.h>
#include <hip/hip_bf16.h>
#include <stdint.h>

// Problem shape (fixed by the reference file)
#define BB   8
#define CC   128
#define HH   256
#define WW   256

#define TH      16              // output rows per block tile
#define NROWS   (TH + 2)        // staged input rows (with top/bottom halo)
#define LDSW    (WW + 8)        // LDS row stride in floats (264): keeps 16B alignment
#define DOFF    4               // data starts at column 4 of each LDS row (16B aligned)
#define CHUNKS  (WW / 4)        // 64 x b128 chunks per row
#define NTHREADS 256
#define NVEC4   ((NROWS * LDSW) / 4)   // 1188 float4s in the tile (LDSW%4==0)

// gfx1250 async global->LDS copy path (ASYNCcnt-tracked), with safe fallback.
#if defined(__gfx1250__)
#  if __has_builtin(__builtin_amdgcn_global_load_async_to_lds_b128)
#    define USE_ASYNC_LDS 1
#  endif
#endif
#ifndef USE_ASYNC_LDS
#  define USE_ASYNC_LDS 0
#endif

#if USE_ASYNC_LDS
// Builtin parameter types (confirmed via hipcc diagnostic in an earlier round):
//   param0: __attribute__((vector_size(16))) int  addrspace(1)*   (global src)
//   param1: same pointee, addrspace(3)                            (LDS dst)
typedef int v4i_gcc __attribute__((vector_size(16)));
typedef __attribute__((address_space(1))) v4i_gcc* async_gptr_t;
typedef __attribute__((address_space(3))) v4i_gcc* async_lptr_t;
#endif

__global__ __launch_bounds__(NTHREADS) void hex_dwconv_kernel(
    const float* __restrict__ x,
    const float* __restrict__ k0,   // [C][3]
    const float* __restrict__ k1,   // [C][2][2]
    const float* __restrict__ bias, // [C]
    float* __restrict__ out)
{
    __shared__ __align__(16) float tile[NROWS * LDSW];

    const int tid      = threadIdx.x;
    const int tiles_pp = HH / TH;                  // 16 row-tiles per plane
    const int plane    = blockIdx.x / tiles_pp;    // (b*C + ch)
    const int tile_id  = blockIdx.x % tiles_pp;
    const int r0       = tile_id * TH;
    const int ch       = plane % CC;

    const float* xplane = x + (size_t)plane * (HH * WW);
    float*       oplane = out + (size_t)plane * (HH * WW);

    // ---- per-channel weights (block-uniform -> scalar loads) ----
    const float w00 = k0[ch * 3 + 0];
    const float w01 = k0[ch * 3 + 1];
    const float w02 = k0[ch * 3 + 2];
    const float s00 = k1[ch * 4 + 0];   // k1[ch][0][0]
    const float s01 = k1[ch * 4 + 1];   // k1[ch][0][1]
    const float s10 = k1[ch * 4 + 2];   // k1[ch][1][0]
    const float s11 = k1[ch * 4 + 3];   // k1[ch][1][1]
    const float bv  = bias[ch];

    // ---- phase 1: zero LDS tile with b128 stores (halo cols / OOB halo rows) ----
    {
        float4* t4 = (float4*)tile;
        const float4 z = make_float4(0.f, 0.f, 0.f, 0.f);
        #pragma unroll
        for (int i = tid; i < NVEC4; i += NTHREADS)
            t4[i] = z;
    }
    __syncthreads();   // ds stores drained before async engine writes LDS

    // ---- phase 2: stage NROWS x WW floats into LDS with async b128 copies ----
    for (int idx = tid; idx < NROWS * CHUNKS; idx += NTHREADS) {
        const int ri = idx >> 6;        // row within tile   (0..17)
        const int ck = idx & 63;        // b128 chunk in row (0..63)
        const int gr = r0 - 1 + ri;     // global input row
        if (gr >= 0 && gr < HH) {
            const float* gp = xplane + gr * WW + ck * 4;
            float*       lp = &tile[ri * LDSW + DOFF + ck * 4];
#if USE_ASYNC_LDS
            __builtin_amdgcn_global_load_async_to_lds_b128(
                (async_gptr_t)gp, (async_lptr_t)lp, 0, 0);
#else
            const float4 v = *(const float4*)gp;
            *(float4*)lp = v;
#endif
        }
    }
#if USE_ASYNC_LDS
#  if __has_builtin(__builtin_amdgcn_s_wait_asynccnt)
    __builtin_amdgcn_s_wait_asynccnt(0);
#  else
    asm volatile("s_wait_asynccnt 0" ::: "memory");
#  endif
#endif
    __syncthreads();

    // ---- phase 3: compute; one thread per output column, roll down TH rows ----
    const int c  = tid;                 // output column 0..255
    const int lc = DOFF + c;            // LDS column of center tap
    const bool even = ((c & 1) == 0);

    // rolling 3x3 register window: t=row r-1, m=row r (relative to output row)
    float tl = tile[0 * LDSW + lc - 1];
    float tc = tile[0 * LDSW + lc];
    float tr = tile[0 * LDSW + lc + 1];
    float ml = tile[1 * LDSW + lc - 1];
    float mc = tile[1 * LDSW + lc];
    float mr = tile[1 * LDSW + lc + 1];

    #pragma unroll
    for (int ri = 0; ri < TH; ++ri) {
        const float* brow = &tile[(ri + 2) * LDSW + lc];
        const float bl = brow[-1];
        const float bc = brow[0];
        const float br = brow[1];

        // center column: rows r-1, r, r+1
        float acc = fmaf(w00, tc, fmaf(w01, mc, fmaf(w02, bc, bv)));

        // side columns: even c -> rows (r, r+1); odd c -> rows (r-1, r)
        const float al  = even ? ml : tl;
        const float bl2 = even ? bl : ml;
        const float ar  = even ? mr : tr;
        const float br2 = even ? br : mr;
        acc = fmaf(s00, al, acc);
        acc = fmaf(s10, bl2, acc);
        acc = fmaf(s01, ar, acc);
        acc = fmaf(s11, br2, acc);

        oplane[(r0 + ri) * WW + c] = acc;

        tl = ml; tc = mc; tr = mr;      // shift window down one row
        ml = bl; mc = bc; mr = br;
    }
}

extern "C" void kernel_launch(void* const* d_in, const int* in_sizes, int n_in,
                              void* d_out, int out_size, void* d_ws, size_t ws_size,
                              hipStream_t stream) {
    const float* x  = (const float*)d_in[0];
    const float* k0 = (const float*)d_in[1];
    const float* k1 = (const float*)d_in[2];
    const float* b  = (const float*)d_in[3];
    float* out = (float*)d_out;

    const int nblocks = BB * CC * (HH / TH);   // 8*128*16 = 16384
    hex_dwconv_kernel<<<nblocks, NTHREADS, 0, stream>>>(x, k0, k1, b, out);
}